// GCNNormNodeLabelAggregator_16535624090045
// MI455X (gfx1250) — compile-verified
//
#include <hip/hip_runtime.h>
#include <hip/hip_bf16.h>
#include <stdint.h>

// GCN-norm neighbor aggregation: out = concat(x, A_norm @ x), A from edge_index.
// Bandwidth-bound (0.25 FLOP/byte) -> no WMMA (random sparsity, no dense tiles).
// Strategy: counting-sort edges into CSR (u32 atomics only), then atomic-free
// register-accumulating pull fused with the x-copy (full 512B/row stores).
// CDNA5 paths: global_load_async_to_lds_b128 + s_wait_asynccnt edge staging,
// global_prefetch_b8 in the gather loop, native u32 global atomics, v_rsq_f32,
// NT hints. Working set ~100 MB fits the 192 MB L2.

typedef float v4f __attribute__((ext_vector_type(4)));

#define FEAT  64
#define TWOF  128
#define BLK   256
#define CHUNK 512   // edges staged per block (2*CHUNK int64 = 8 KB LDS)

// ---- CDNA5 async load (global -> LDS), tracked by ASYNCcnt --------------------
__device__ __forceinline__ void async_load_b128_to_lds(uint32_t lds_byte_off,
                                                       const void* gaddr) {
    asm volatile("global_load_async_to_lds_b128 %0, %1, off"
                 :: "v"(lds_byte_off),
                    "v"((uint64_t)(uintptr_t)gaddr)
                 : "memory");
}
__device__ __forceinline__ void wait_asynccnt0() {
    asm volatile("s_wait_asynccnt 0" ::: "memory");
}

// ---- CSR scatter: edge tiles async-staged to LDS; cursor atomics place cols ---
__global__ void k_scatter(const long long* __restrict__ ei,
                          uint32_t* __restrict__ cursor,
                          int* __restrict__ ecol, int E) {
    __shared__ long long sidx[2 * CHUNK];   // [0,CHUNK) rows, [CHUNK,2*CHUNK) cols

    const int e0 = blockIdx.x * CHUNK;
    const uint32_t lds_base = (uint32_t)(uintptr_t)&sidx[0]; // low 32b = LDS offset
    const char* grow = (const char*)(ei + (size_t)e0);
    const char* gcol = (const char*)(ei + (size_t)E + e0);

    // 8192 B = 512 x 16 B chunks; 256 threads issue 2 async b128 loads each.
    #pragma unroll
    for (int k = 0; k < 2; ++k) {
        int c = (int)threadIdx.x + k * BLK;                 // 0..511
        const char* src = (c < 256) ? (grow + (size_t)c * 16)
                                    : (gcol + (size_t)(c - 256) * 16);
        async_load_b128_to_lds(lds_base + (uint32_t)c * 16u, src);
    }
    wait_asynccnt0();
    __syncthreads();

    #pragma unroll
    for (int k = 0; k < 2; ++k) {
        int eL  = (int)threadIdx.x + k * BLK;               // 0..511
        int row = (int)sidx[eL];
        int col = (int)sidx[CHUNK + eL];
        uint32_t p = atomicAdd(&cursor[row], 1u);           // global_atomic_add_u32
        ecol[p] = col;
    }
}

__global__ void k_scatter_tail(const long long* __restrict__ ei,
                               uint32_t* __restrict__ cursor,
                               int* __restrict__ ecol, int E, int e_start) {
    int e = e_start + blockIdx.x * blockDim.x + threadIdx.x;
    if (e < E) {
        int row = (int)ei[e];
        int col = (int)ei[(size_t)E + e];
        uint32_t p = atomicAdd(&cursor[row], 1u);
        ecol[p] = col;
    }
}

// ---- fused copy + atomic-free pull: 16 lanes/row, full 512B/row stores --------
__global__ void k_fused(const uint32_t* __restrict__ start,
                        const uint32_t* __restrict__ deg,
                        const int* __restrict__ ecol,
                        const float* __restrict__ x,
                        const float* __restrict__ dis,
                        float* __restrict__ out, int N) {
    int t = blockIdx.x * blockDim.x + threadIdx.x;
    int n = t >> 4, l = t & 15;
    if (n >= N) return;

    // out[n, :64] = x[n]  (contiguous with the second-half store below)
    v4f vx = *((const v4f*)(x + (size_t)n * FEAT) + l);
    __builtin_nontemporal_store(vx, (v4f*)(out + (size_t)n * TWOF) + l);

    // out[n, 64:] = sum_j dis[n]*dis[col_j] * x[col_j]
    uint32_t s = start[n];
    uint32_t d = deg[n];
    float   wr = dis[n];
    v4f acc = {0.0f, 0.0f, 0.0f, 0.0f};
    for (uint32_t j = 0; j < d; ++j) {
        int col = ecol[s + j];                                // broadcast load
        if (j + 1 < d) {                                      // hide gather latency
            int cn = ecol[s + j + 1];
            __builtin_prefetch(x + (size_t)cn * FEAT + l * 4, 0, 3); // global_prefetch_b8
        }
        float w = wr * dis[col];
        v4f v = *((const v4f*)(x + (size_t)col * FEAT) + l);  // 256 B burst, L2 hit
        acc += v * w;                                         // v_fmac_f32 x4
    }
    *((v4f*)(out + (size_t)n * TWOF + FEAT) + l) = acc;       // no atomics
}

// ---- degree count / normalization --------------------------------------------
__global__ void k_zero_deg(uint32_t* __restrict__ deg, int N) {
    int i = blockIdx.x * blockDim.x + threadIdx.x;
    if (i < N) deg[i] = 0u;
}

__global__ void k_count(const long long* __restrict__ ei,
                        uint32_t* __restrict__ deg, int E) {
    int e = blockIdx.x * blockDim.x + threadIdx.x;
    if (e < E) {
        int row = (int)__builtin_nontemporal_load(&ei[e]);
        atomicAdd(&deg[row], 1u);
    }
}

__global__ void k_dis(const uint32_t* __restrict__ deg,
                      float* __restrict__ dis, int N) {
    int i = blockIdx.x * blockDim.x + threadIdx.x;
    if (i < N) {
        uint32_t d = deg[i];
        dis[i] = (d > 0u) ? rsqrtf((float)d) : 0.0f;          // v_rsq_f32
    }
}

// ---- exclusive scan over deg[] (3 kernels, wave32 shuffle scan) ---------------
__global__ void k_scan1(const uint32_t* __restrict__ deg,
                        uint32_t* __restrict__ start,
                        uint32_t* __restrict__ blksum, int N) {
    int i = blockIdx.x * BLK + threadIdx.x;
    unsigned orig = (i < N) ? deg[i] : 0u;
    unsigned v = orig;
    int lane = threadIdx.x & 31, wid = threadIdx.x >> 5;
    #pragma unroll
    for (int off = 1; off < 32; off <<= 1) {
        unsigned tv = __shfl_up(v, off);                      // wave32 scan
        if (lane >= off) v += tv;
    }
    __shared__ unsigned wsum[BLK / 32];
    if (lane == 31) wsum[wid] = v;
    __syncthreads();
    unsigned wadd = 0;
    for (int k = 0; k < wid; ++k) wadd += wsum[k];            // <=7 LDS broadcasts
    unsigned incl = v + wadd;
    if (i < N) start[i] = incl - orig;                        // block-local exclusive
    if (threadIdx.x == BLK - 1) blksum[blockIdx.x] = incl;    // block total
}

__global__ void k_scan2(const uint32_t* __restrict__ blksum,
                        uint32_t* __restrict__ blkoff, int nb) {
    // single block of 512 threads; nb <= 512 (N <= 131072)
    int i = threadIdx.x;
    unsigned orig = (i < nb) ? blksum[i] : 0u;
    unsigned v = orig;
    int lane = threadIdx.x & 31, wid = threadIdx.x >> 5;
    #pragma unroll
    for (int off = 1; off < 32; off <<= 1) {
        unsigned tv = __shfl_up(v, off);
        if (lane >= off) v += tv;
    }
    __shared__ unsigned wsum[512 / 32];
    if (lane == 31) wsum[wid] = v;
    __syncthreads();
    unsigned wadd = 0;
    for (int k = 0; k < wid; ++k) wadd += wsum[k];
    unsigned incl = v + wadd;
    if (i < nb) blkoff[i] = incl - orig;                      // exclusive
}

__global__ void k_scan3(uint32_t* __restrict__ start,
                        uint32_t* __restrict__ cursor,
                        const uint32_t* __restrict__ blkoff, int N) {
    int i = blockIdx.x * BLK + threadIdx.x;
    if (i < N) {
        unsigned s = start[i] + blkoff[i >> 8];
        start[i]  = s;
        cursor[i] = s;
    }
}

// ---- fallback path pieces (used only if ws too small for CSR) -----------------
__global__ void k_copy(const float* __restrict__ x, float* __restrict__ out, int N) {
    int t = blockIdx.x * blockDim.x + threadIdx.x;
    int n = t >> 4, l = t & 15;
    if (n < N) {
        v4f v = *((const v4f*)(x + (size_t)n * FEAT) + l);
        __builtin_nontemporal_store(v, (v4f*)(out + (size_t)n * TWOF) + l);
        v4f z = {0.0f, 0.0f, 0.0f, 0.0f};
        *((v4f*)(out + (size_t)n * TWOF + FEAT) + l) = z;
    }
}

__global__ void k_agg_push(const long long* __restrict__ ei,
                           const float* __restrict__ x,
                           const float* __restrict__ dis,
                           float* __restrict__ out, int E) {
    int t = blockIdx.x * blockDim.x + threadIdx.x;
    int e = t >> 4, l = t & 15;
    if (e < E) {
        int row = (int)ei[e];
        int col = (int)ei[(size_t)E + e];
        float w = dis[row] * dis[col];
        v4f v = *((const v4f*)(x + (size_t)col * FEAT) + l);
        float* o = out + (size_t)row * TWOF + FEAT + l * 4;
        unsafeAtomicAdd(o + 0, w * v.x);                      // global_atomic_add_f32
        unsafeAtomicAdd(o + 1, w * v.y);
        unsafeAtomicAdd(o + 2, w * v.z);
        unsafeAtomicAdd(o + 3, w * v.w);
    }
}

extern "C" void kernel_launch(void* const* d_in, const int* in_sizes, int n_in,
                              void* d_out, int out_size, void* d_ws, size_t ws_size,
                              hipStream_t stream) {
    const float*     x   = (const float*)d_in[0];
    const long long* ei  = (const long long*)d_in[1];
    float*           out = (float*)d_out;

    const int N  = in_sizes[0] / FEAT;  // 100000
    const int E  = in_sizes[1] / 2;     // 1200000
    const int gN = (N + BLK - 1) / BLK;
    const int nb = gN;                  // scan1 block count

    // ws layout
    uint32_t* deg    = (uint32_t*)d_ws;              // N
    float*    dis    = (float*)(deg + N);            // N
    uint32_t* start  = (uint32_t*)(dis + N);         // N
    uint32_t* cursor = start + N;                    // N
    int*      ecol   = (int*)(cursor + N);           // E
    uint32_t* blksum = (uint32_t*)(ecol + E);        // <=512
    uint32_t* blkoff = blksum + 512;                 // <=512
    const size_t need_csr = (size_t)16 * N + (size_t)4 * E + 4096;

    // degree + normalization (both paths need these)
    k_zero_deg<<<gN, BLK, 0, stream>>>(deg, N);
    k_count<<<(E + BLK - 1) / BLK, BLK, 0, stream>>>(ei, deg, E);
    k_dis<<<gN, BLK, 0, stream>>>(deg, dis, N);

    if (ws_size >= need_csr && nb <= 512) {
        // ---- CSR pull path (no fp atomics) ----
        k_scan1<<<nb, BLK, 0, stream>>>(deg, start, blksum, N);
        k_scan2<<<1, 512, 0, stream>>>(blksum, blkoff, nb);
        k_scan3<<<gN, BLK, 0, stream>>>(start, cursor, blkoff, N);

        const int full = E / CHUNK;
        if (full > 0)
            k_scatter<<<full, BLK, 0, stream>>>(ei, cursor, ecol, E);
        const int e_start = full * CHUNK;
        if (E - e_start > 0)
            k_scatter_tail<<<(E - e_start + BLK - 1) / BLK, BLK, 0, stream>>>(
                ei, cursor, ecol, E, e_start);

        k_fused<<<((size_t)N * 16 + BLK - 1) / BLK, BLK, 0, stream>>>(
            start, deg, ecol, x, dis, out, N);
    } else {
        // ---- fallback push path (f32 atomics) ----
        k_copy<<<((size_t)N * 16 + BLK - 1) / BLK, BLK, 0, stream>>>(x, out, N);
        k_agg_push<<<((size_t)E * 16 + BLK - 1) / BLK, BLK, 0, stream>>>(
            ei, x, dis, out, E);
    }
}